// STDLayer_8126078124652
// MI455X (gfx1250) — compile-verified
//
#include <hip/hip_runtime.h>

typedef float v2f __attribute__((ext_vector_type(2)));
typedef float v8f __attribute__((ext_vector_type(8)));

#define NCH   21
#define IMG   512
#define HWSZ  (IMG * IMG)
#define HALFK 3
#define TILE  16
#define HALO  22           // TILE + 2*HALFK
#define TPB   256
#define NWAVE 8
#define WPAD  40           // padded banded tap row: index m+15, m in [-15, 24]

// ---------------------------------------------------------------------------
// Initial softmax: u0 = softmax(o * eps^2, axis=C). One thread per pixel.
// ---------------------------------------------------------------------------
__global__ __launch_bounds__(TPB) void std_init_softmax(
    const float* __restrict__ o, const float* __restrict__ eps_p,
    float* __restrict__ u_out)
{
    size_t pix = (size_t)blockIdx.x * blockDim.x + threadIdx.x; // over B*H*W
    if (pix >= (size_t)8 * HWSZ) return;
    size_t b = pix / HWSZ;
    size_t p = pix % HWSZ;
    size_t base = b * (size_t)NCH * HWSZ + p;
    float eps2 = eps_p[0] * eps_p[0];

    float s[NCH];
    float mx = -3.4e38f;
#pragma unroll
    for (int c = 0; c < NCH; ++c) {
        float v = o[base + (size_t)c * HWSZ] * eps2;
        s[c] = v;
        mx = fmaxf(mx, v);
    }
    float sum = 0.f;
#pragma unroll
    for (int c = 0; c < NCH; ++c) { float e = __expf(s[c] - mx); s[c] = e; sum += e; }
    float r = 1.f / sum;
#pragma unroll
    for (int c = 0; c < NCH; ++c) u_out[base + (size_t)c * HWSZ] = s[c] * r;
}

// ---------------------------------------------------------------------------
// Fused step: q = depthwise 7x7 Gaussian conv of (1 - 2u)  (separable, zero-pad)
//             u_new = softmax((o - lam*q) * eps^2, axis=C)
// One workgroup = one 16x16 spatial tile of one batch image, all 21 channels.
// Vertical conv pass via V_WMMA_F32_16X16X4_F32 (6 chained K=4 chunks):
//   Q(16x16) = Wv(16x24) x H(24x16), Wv banded with the 1D taps.
// A-operand elements come from a zero-padded banded tap table in LDS, so no
// select chains are needed (index m+15 is always in [0, 39]).
// ---------------------------------------------------------------------------
__global__ __launch_bounds__(TPB) void std_step_kernel(
    const float* __restrict__ o, const float* __restrict__ u_in,
    const float* __restrict__ nb_sigma, const float* __restrict__ eps_p,
    const float* __restrict__ lam_p, float* __restrict__ u_out)
{
    __shared__ float wtab[NCH][WPAD];               // banded 1D taps, zero-padded
    __shared__ float t_slab[NWAVE][HALO][HALO + 1]; // per-wave input tile
    __shared__ float q_tile[NCH][TILE][TILE];       // conv results for softmax

    const int tid  = threadIdx.x;
    const int wave = tid >> 5;     // wave32
    const int lane = tid & 31;
    const int half = lane >> 4;    // 0: lanes 0-15, 1: lanes 16-31
    const int n    = lane & 15;    // A: output row M ; B/D: output col N

    const int tiles_x       = IMG / TILE;            // 32
    const int tiles_per_img = tiles_x * tiles_x;     // 1024
    const int b  = blockIdx.x / tiles_per_img;
    const int t2 = blockIdx.x % tiles_per_img;
    const int y0 = (t2 / tiles_x) * TILE;
    const int x0 = (t2 % tiles_x) * TILE;

    const float eps2 = eps_p[0] * eps_p[0];
    const float lam  = lam_p[0];

    // --- fill the banded tap table (one pass, zero outside band) -----------
    for (int idx = tid; idx < NCH * WPAD; idx += TPB) {
        int c   = idx / WPAD;
        int m15 = idx % WPAD;
        float v = 0.f;
        int k = m15 - 15;                      // tap index
        if (k >= 0 && k < 7) {
            float sig = nb_sigma[c];
            float inv2s2 = 1.f / (2.f * sig * sig);
            float s = 0.f, g = 0.f;
#pragma unroll
            for (int j = 0; j < 7; ++j) {
                float d = (float)(j - HALFK);
                float e = __expf(-(d * d) * inv2s2);
                s += e;
                if (j == k) g = e;
            }
            v = g / s;                         // normalized separable 1D tap
        }
        wtab[c][m15] = v;
    }
    __syncthreads();

    float (*tt)[HALO + 1] = t_slab[wave];

    for (int c = wave; c < NCH; c += NWAVE) {
        // taps for the horizontal pass (LDS broadcast reads)
        float w[7];
#pragma unroll
        for (int k = 0; k < 7; ++k) w[k] = wtab[c][15 + k];

        // --- stage 22x22 tile of t = 1 - 2u into LDS (zero outside image) --
        const float* up = u_in + ((size_t)b * NCH + c) * HWSZ;
        for (int idx = lane; idx < HALO * HALO; idx += 32) {
            int r  = idx / HALO;
            int cc = idx % HALO;
            int gy = y0 - HALFK + r;
            int gx = x0 - HALFK + cc;
            float v = 0.f;
            if (gy >= 0 && gy < IMG && gx >= 0 && gx < IMG)
                v = 1.f - 2.f * up[gy * IMG + gx];
            tt[r][cc] = v;
        }
        asm volatile("s_wait_dscnt 0" ::: "memory"); // wave-local LDS RAW

        // --- separable conv: horizontal on VALU, vertical via WMMA ---------
        v8f acc = {};
#pragma unroll
        for (int c6 = 0; c6 < 6; ++c6) {
            int r0 = 4 * c6 + 2 * half;     // this lane's two K indices: r0, r0+1
            v2f avec, bvec;
#pragma unroll
            for (int j = 0; j < 2; ++j) {
                int r = r0 + j;             // H row (halo coords)
                // B operand: H[r][n] = horizontal 7-tap at halo row r, out col n
                float h = 0.f;
                if (r < HALO) {
#pragma unroll
                    for (int k = 0; k < 7; ++k) h += w[k] * tt[r][n + k];
                }
                bvec[j] = h;
                // A operand: Wv[n][r] = w1d[r - n], banded table lookup
                avec[j] = wtab[c][r - n + 15];
            }
            // D = A x B + C  (neg_a, A, neg_b, B, c_mod, C, reuse_a, reuse_b)
            acc = __builtin_amdgcn_wmma_f32_16x16x4_f32(
                false, avec, false, bvec, (short)0, acc, false, false);
        }

        // D layout: VGPR g -> row g + 8*half, col n
#pragma unroll
        for (int g = 0; g < 8; ++g)
            q_tile[c][g + 8 * half][n] = acc[g];
    }

    // prefetch o operand for the softmax phase while conv results settle
    {
        int y = tid >> 4, x = tid & 15;
        const float* op = o + (size_t)b * NCH * HWSZ + (size_t)(y0 + y) * IMG + (x0 + x);
        __builtin_prefetch(op, 0, 1);
        __builtin_prefetch(op + (size_t)10 * HWSZ, 0, 1);
        __builtin_prefetch(op + (size_t)20 * HWSZ, 0, 1);
    }

    __syncthreads();

    // --- channel softmax per pixel: 256 threads == 16x16 pixels ------------
    {
        int y = tid >> 4, x = tid & 15;
        size_t base = (size_t)b * NCH * HWSZ + (size_t)(y0 + y) * IMG + (x0 + x);
        float s[NCH];
        float mx = -3.4e38f;
#pragma unroll
        for (int c = 0; c < NCH; ++c) {
            float q = q_tile[c][y][x];
            float v = (o[base + (size_t)c * HWSZ] - lam * q) * eps2;
            s[c] = v;
            mx = fmaxf(mx, v);
        }
        float sum = 0.f;
#pragma unroll
        for (int c = 0; c < NCH; ++c) { float e = __expf(s[c] - mx); s[c] = e; sum += e; }
        float r = 1.f / sum;
#pragma unroll
        for (int c = 0; c < NCH; ++c) u_out[base + (size_t)c * HWSZ] = s[c] * r;
    }
}

// ---------------------------------------------------------------------------
// Launch: u0 = softmax(o*eps2) -> d_out, then 10 fused steps ping-ponging
// between d_out and d_ws so the final u lands in d_out.
// ---------------------------------------------------------------------------
extern "C" void kernel_launch(void* const* d_in, const int* in_sizes, int n_in,
                              void* d_out, int out_size, void* d_ws, size_t ws_size,
                              hipStream_t stream) {
    (void)in_sizes; (void)n_in; (void)out_size; (void)ws_size;
    const float* o        = (const float*)d_in[0];
    const float* nb_sigma = (const float*)d_in[1];
    const float* eps      = (const float*)d_in[2];
    const float* lam      = (const float*)d_in[3];

    float* bufs[2] = { (float*)d_out, (float*)d_ws };

    const int npix_blocks = (8 * HWSZ + TPB - 1) / TPB;  // 8192
    std_init_softmax<<<npix_blocks, TPB, 0, stream>>>(o, eps, bufs[0]);

    const int step_blocks = 8 * (IMG / TILE) * (IMG / TILE); // 8192
    for (int s = 1; s <= 10; ++s) {
        const float* uin = bufs[(s + 1) & 1];
        float*       uo  = bufs[s & 1];
        std_step_kernel<<<step_blocks, TPB, 0, stream>>>(o, uin, nb_sigma, eps, lam, uo);
    }
}